// Neighbor_Aggregation_28398323761218
// MI455X (gfx1250) — compile-verified
//
#include <hip/hip_runtime.h>

// Problem constants from the reference.
#define B_  2
#define N_  50000
#define D_  32
#define E_  1600000

#define CHUNK   512          // edges staged per buffer
#define CPB     2            // chunks processed per block
#define NBLOCKS 3125         // 3,200,000 edges = 3125 * 2 * 512 (exact)

// ---------------------------------------------------------------------------
// CDNA5 async global->LDS helpers (ASYNCcnt path).
// the LDS destination address comes from the vdst VGPR; low 32 bits of a generic pointer to LDS
// are the workgroup-relative LDS byte offset.
// ---------------------------------------------------------------------------
__device__ __forceinline__ unsigned lds_off(const void* p) {
    return (unsigned)(uintptr_t)p;
}

__device__ __forceinline__ void async_load_b64(unsigned lds_byte_off, const void* gaddr) {
    asm volatile("global_load_async_to_lds_b64 %0, %1, off"
                 :: "v"(lds_byte_off), "v"(gaddr)
                 : "memory");
}

__device__ __forceinline__ void wait_async_le3() {
    asm volatile("s_wait_asynccnt 3" ::: "memory");
}
__device__ __forceinline__ void wait_async_le0() {
    asm volatile("s_wait_asynccnt 0" ::: "memory");
}

// ---------------------------------------------------------------------------
// Zero-fill d_out (harness poisons it; atomics need a clean accumulator).
// ---------------------------------------------------------------------------
__global__ __launch_bounds__(256) void na_zero_kernel(float4* __restrict__ out, int n4) {
    int i = blockIdx.x * 256 + threadIdx.x;
    if (i < n4) {
        out[i] = float4{0.0f, 0.0f, 0.0f, 0.0f};
    }
}

// ---------------------------------------------------------------------------
// Neighbor aggregation: out[b, src[e], :] += w[e] * H[b, dst[e], :]
//   - edge stream double-buffered into LDS via async global->LDS DMA
//   - 8 threads per edge, each owns one float4 of the D=32 row
//   - gather = coalesced 128B global_load_b128; scatter = native f32 atomics
// ---------------------------------------------------------------------------
__global__ __launch_bounds__(256) void na_agg_kernel(
        const float*  __restrict__ H,        // [B, N, D]
        const int*    __restrict__ edge_src, // [B, E] flat
        const int*    __restrict__ edge_dst, // [B, E] flat
        const float*  __restrict__ edge_w,   // [B, E] flat
        float*        __restrict__ out)      // [B, N, D]
{
    __shared__ __align__(16) int   sh_src[2][CHUNK];
    __shared__ __align__(16) int   sh_dst[2][CHUNK];
    __shared__ __align__(16) float sh_w  [2][CHUNK];

    const unsigned tid    = threadIdx.x;
    const unsigned chunk0 = blockIdx.x * CPB;          // first chunk of this block
    const unsigned q      = tid & 7u;                  // float4 slot within row
    const unsigned eo     = tid >> 3;                  // edge-octet id within step

    // Stage chunk0 into buffer 0: each thread moves 8 bytes (2 elems) per array.
    {
        const unsigned base = chunk0 * CHUNK + tid * 2u;
        async_load_b64(lds_off(&sh_src[0][tid * 2u]), edge_src + base);
        async_load_b64(lds_off(&sh_dst[0][tid * 2u]), edge_dst + base);
        async_load_b64(lds_off(&sh_w  [0][tid * 2u]), edge_w   + base);
    }

    for (unsigned c = 0; c < CPB; ++c) {
        const unsigned buf = c & 1u;

        // Prefetch next chunk into the other buffer (safe: all waves finished
        // reading it at the trailing barrier of iteration c-1).
        if (c + 1u < CPB) {
            const unsigned nbase = (chunk0 + c + 1u) * CHUNK + tid * 2u;
            async_load_b64(lds_off(&sh_src[buf ^ 1u][tid * 2u]), edge_src + nbase);
            async_load_b64(lds_off(&sh_dst[buf ^ 1u][tid * 2u]), edge_dst + nbase);
            async_load_b64(lds_off(&sh_w  [buf ^ 1u][tid * 2u]), edge_w   + nbase);
            wait_async_le3();   // drain the 3 older ops (chunk c); keep 3 in flight
        } else {
            wait_async_le0();   // last chunk: drain everything
        }
        __syncthreads();        // staged data visible to all waves

        // Process chunk c: CHUNK edges, 32 edges per step (8 threads/edge).
        const unsigned cb_edges = (chunk0 + c) * CHUNK;
#pragma unroll 4
        for (unsigned step = 0; step < (CHUNK / 32u); ++step) {
            const unsigned eL = (step << 5) + eo;       // edge index inside chunk
            const int   s = sh_src[buf][eL];
            const int   d = sh_dst[buf][eL];
            const float w = sh_w  [buf][eL];

            const unsigned eg   = cb_edges + eL;        // global edge id in [0, B*E)
            const unsigned boff = (eg >= (unsigned)E_) ? (unsigned)(N_ * D_) : 0u;

            // Gather: 128-bit coalesced load of 4 floats of H[b, d, :].
            const float4 hv = *(const float4*)(H + boff + (unsigned)d * (unsigned)D_ + (q << 2));

            // Scatter-add: 4 native f32 atomics into out[b, s, :].
            float* o = out + boff + (unsigned)s * (unsigned)D_ + (q << 2);
#if defined(__HIP_DEVICE_COMPILE__)
            unsafeAtomicAdd(o + 0, w * hv.x);
            unsafeAtomicAdd(o + 1, w * hv.y);
            unsafeAtomicAdd(o + 2, w * hv.z);
            unsafeAtomicAdd(o + 3, w * hv.w);
#else
            atomicAdd(o + 0, w * hv.x);
            atomicAdd(o + 1, w * hv.y);
            atomicAdd(o + 2, w * hv.z);
            atomicAdd(o + 3, w * hv.w);
#endif
        }
        __syncthreads();        // all waves done reading buf before it is re-staged
    }
}

// ---------------------------------------------------------------------------
// Launch
// ---------------------------------------------------------------------------
extern "C" void kernel_launch(void* const* d_in, const int* in_sizes, int n_in,
                              void* d_out, int out_size, void* d_ws, size_t ws_size,
                              hipStream_t stream) {
    const float* H        = (const float*)d_in[0];
    const int*   edge_src = (const int*)  d_in[1];
    const int*   edge_dst = (const int*)  d_in[2];
    const float* edge_w   = (const float*)d_in[3];
    float*       out      = (float*)      d_out;

    // 1) Zero the accumulator (out_size = B*N*D = 3,200,000 floats).
    const int n4 = out_size / 4;
    const int zb = (n4 + 255) / 256;
    na_zero_kernel<<<zb, 256, 0, stream>>>((float4*)out, n4);

    // 2) Aggregate: 3125 blocks x 256 threads, each block handles 2x512 edges.
    na_agg_kernel<<<NBLOCKS, 256, 0, stream>>>(H, edge_src, edge_dst, edge_w, out);
}